// MatchingAttention_25091198943368
// MI455X (gfx1250) — compile-verified
//
#include <hip/hip_runtime.h>
#include <hip/hip_bf16.h>
#include <math.h>

// Problem constants (from reference): S=256, B=512, D=1024, CAND=1024
#define SEQ   256
#define BATCH 512
#define DIM   1024
#define CAND  1024

typedef float v2f __attribute__((ext_vector_type(2)));
typedef float v8f __attribute__((ext_vector_type(8)));

// ---------------------------------------------------------------------------
// Kernel 1: x_ = x @ W^T + b   (B x D) via V_WMMA_F32_16X16X4_F32
// One wave (32 threads) computes one 16x16 output tile, K-loop over CAND in
// steps of 4. A-tile = rows of x (batch-major), B-tile = W^T => per-lane loads
// are identical row-offset patterns on x and W. f32 WMMA keeps full precision.
// ---------------------------------------------------------------------------
__global__ void __launch_bounds__(32)
gemm_xWT_bias_wmma(const float* __restrict__ x,   // (B, CAND)
                   const float* __restrict__ W,   // (D, CAND)
                   const float* __restrict__ bias,// (D)
                   float* __restrict__ xt)        // (B, D) out
{
    const int tiles_n = DIM / 16;                 // 64 column tiles
    const int tm = blockIdx.x / tiles_n;          // batch tile 0..31
    const int tn = blockIdx.x % tiles_n;          // dim tile   0..63
    const int lane = threadIdx.x;                 // 0..31 (wave32)
    const int half = lane >> 4;                   // 0: lanes 0-15, 1: lanes 16-31
    const int l16  = lane & 15;

    // A: 16x4 f32, lane(l16)=row M; VGPR0/1 hold K = 2*half + {0,1}
    // B: 4x16 f32, lane(l16)=col N; VGPR0/1 hold K = 2*half + {0,1}; B[k][n]=W[n,k]
    const float* __restrict__ xrow = x + (size_t)(tm * 16 + l16) * CAND;
    const float* __restrict__ wrow = W + (size_t)(tn * 16 + l16) * CAND;

    v8f acc = {};
#pragma unroll 4
    for (int k = 0; k < CAND; k += 4) {
        const int ko = k + half * 2;              // 8-byte aligned (ko even)
        v2f a = *(const v2f*)(xrow + ko);
        v2f bmat = *(const v2f*)(wrow + ko);
        acc = __builtin_amdgcn_wmma_f32_16x16x4_f32(
            /*neg_a=*/false, a, /*neg_b=*/false, bmat,
            /*c_mod=*/(short)0, acc, /*reuse_a=*/false, /*reuse_b=*/false);
    }

    // C/D layout: VGPR j -> M = j + 8*half (lanes 0-15: M=j, lanes 16-31: M=j+8)
    const int n = tn * 16 + l16;
    const float bn = bias[n];
#pragma unroll
    for (int j = 0; j < 8; ++j) {
        const int m = tm * 16 + j + half * 8;
        xt[(size_t)m * DIM + n] = acc[j] + bn;
    }
}

// ---------------------------------------------------------------------------
// Kernel 2: one block per batch b (256 threads = 8 waves).
//   Phase A: scores[s] = tanh(mask^2 * dot(x_[b,:], M[s,b,:])), wave-per-s,
//            coalesced b128 streaming of M rows, x_[b,:] staged in LDS.
//   Phase B: masked softmax + renorm entirely in LDS (alpha written out).
//   Phase C: pool[b,d] = sum_s alpha[s]*M[s,b,d], thread-per-4d float4,
//            second coalesced streaming pass over M.
// ---------------------------------------------------------------------------
__global__ void __launch_bounds__(256)
matching_attention_kernel(const float* __restrict__ M,    // (S, B, D)
                          const float* __restrict__ mask, // (B, S)
                          const float* __restrict__ xt,   // (B, D)
                          float* __restrict__ pool,       // (B, D) out
                          float* __restrict__ alpha_out)  // (B, S) out
{
    const int b    = blockIdx.x;
    const int tid  = threadIdx.x;
    const int lane = tid & 31;
    const int wave = tid >> 5;                    // 0..7

    __shared__ float sh_x[DIM];                   // x_[b,:]   (4 KiB)
    __shared__ float sh_s[SEQ];                   // scores -> e -> alpha
    __shared__ float sh_red[8];
    __shared__ float sh_scalar;                   // broadcast max / sum

    // stage x_[b,:] into LDS (float4, coalesced)
    {
        const float4* src = (const float4*)(xt + (size_t)b * DIM);
        float4* dst = (float4*)sh_x;
        dst[tid] = src[tid];                      // 256 threads * 16B = 4 KiB
    }
    __syncthreads();

    // ---- Phase A: scores, one wave per s ----
    for (int s = wave; s < SEQ; s += 8) {
        const float* __restrict__ row = M + ((size_t)s * BATCH + b) * DIM;
        float4 a4 = make_float4(0.f, 0.f, 0.f, 0.f);
#pragma unroll
        for (int it = 0; it < DIM / 128; ++it) {  // 8 iters, 512B/wave/iter
            const int d = it * 128 + lane * 4;
            float4 mv = *(const float4*)(row + d);
            float4 xv = *(const float4*)(sh_x + d);
            a4.x += mv.x * xv.x; a4.y += mv.y * xv.y;
            a4.z += mv.z * xv.z; a4.w += mv.w * xv.w;
        }
        float acc = (a4.x + a4.y) + (a4.z + a4.w);
#pragma unroll
        for (int off = 16; off > 0; off >>= 1)
            acc += __shfl_xor(acc, off, 32);
        if (lane == 0) {
            const float mk = mask[(size_t)b * SEQ + s];
            sh_s[s] = tanhf(acc * mk * mk);       // masked score -> tanh
        }
    }
    __syncthreads();

    // ---- Phase B: softmax over S, re-mask, renormalize ----
    const float v = sh_s[tid];                    // one score per thread
    // block max
    float mx = v;
#pragma unroll
    for (int off = 16; off > 0; off >>= 1)
        mx = fmaxf(mx, __shfl_xor(mx, off, 32));
    if (lane == 0) sh_red[wave] = mx;
    __syncthreads();
    if (wave == 0) {
        float t = (lane < 8) ? sh_red[lane] : -INFINITY;
#pragma unroll
        for (int off = 4; off > 0; off >>= 1)
            t = fmaxf(t, __shfl_xor(t, off, 32));
        if (lane == 0) sh_scalar = t;
    }
    __syncthreads();
    const float gmax = sh_scalar;

    // e_s = exp(v - max) * mask_s  (softmax denom cancels under renorm)
    const float mk = mask[(size_t)b * SEQ + tid];
    const float e  = __expf(v - gmax) * mk;
    // block sum
    float sm = e;
#pragma unroll
    for (int off = 16; off > 0; off >>= 1)
        sm += __shfl_xor(sm, off, 32);
    if (lane == 0) sh_red[wave] = sm;
    __syncthreads();
    if (wave == 0) {
        float t = (lane < 8) ? sh_red[lane] : 0.f;
#pragma unroll
        for (int off = 4; off > 0; off >>= 1)
            t += __shfl_xor(t, off, 32);
        if (lane == 0) sh_scalar = t;
    }
    __syncthreads();
    const float alpha = e / sh_scalar;
    sh_s[tid] = alpha;
    alpha_out[(size_t)b * SEQ + tid] = alpha;
    __syncthreads();

    // ---- Phase C: attention pooling, thread owns 4 dims ----
    const int d0 = tid * 4;
    float4 acc4 = make_float4(0.f, 0.f, 0.f, 0.f);
    const float* __restrict__ base = M + (size_t)b * DIM + d0;
#pragma unroll 4
    for (int s = 0; s < SEQ; ++s) {
        const float a = sh_s[s];                  // LDS broadcast
        const float4 mv = *(const float4*)(base + (size_t)s * (BATCH * DIM));
        acc4.x += a * mv.x; acc4.y += a * mv.y;
        acc4.z += a * mv.z; acc4.w += a * mv.w;
    }
    *(float4*)(pool + (size_t)b * DIM + d0) = acc4;
}

// ---------------------------------------------------------------------------
// Host-side launcher
// ---------------------------------------------------------------------------
extern "C" void kernel_launch(void* const* d_in, const int* in_sizes, int n_in,
                              void* d_out, int out_size, void* d_ws, size_t ws_size,
                              hipStream_t stream) {
    const float* M    = (const float*)d_in[0];   // (S,B,D)
    const float* x    = (const float*)d_in[1];   // (B,CAND)
    const float* mask = (const float*)d_in[2];   // (B,S)
    const float* W    = (const float*)d_in[3];   // (D,CAND)
    const float* bias = (const float*)d_in[4];   // (D)

    float* pool      = (float*)d_out;                    // (B,D)
    float* alpha_out = (float*)d_out + (size_t)BATCH * DIM; // (B,1,S)
    float* xt        = (float*)d_ws;                     // (B,D) scratch, 2 MiB

    // Phase 1: WMMA GEMM, one wave per 16x16 tile
    const int tiles = (BATCH / 16) * (DIM / 16);         // 2048 waves
    gemm_xWT_bias_wmma<<<tiles, 32, 0, stream>>>(x, W, bias, xt);

    // Phase 2+3: fused scores/softmax/pool, one block per batch
    matching_attention_kernel<<<BATCH, 256, 0, stream>>>(M, mask, xt, pool, alpha_out);
}